// cross_attn_29961691857360
// MI455X (gfx1250) — compile-verified
//
#include <hip/hip_runtime.h>

typedef __attribute__((ext_vector_type(2))) float v2f;
typedef __attribute__((ext_vector_type(8))) float v8f;
typedef __attribute__((ext_vector_type(4))) unsigned int v4u;
typedef __attribute__((ext_vector_type(8))) int v8i;
typedef __attribute__((ext_vector_type(4))) int v4i;

#define DIM 256
#define PCOLS 64                      // weight panel columns per TDM transfer
#define PSTRIDE 68                    // LDS row stride in floats (TDM pads 4 dw per 64 dw)
#define PANEL_BYTES (DIM * PSTRIDE * 4)   // 69632 bytes per panel
#define RED_OFF (2 * PANEL_BYTES)

__device__ __forceinline__ v8f wmma_f32(v2f a, v2f b, v8f c) {
  // D = A(16x4) * B(4x16) + C(16x16), all f32, wave32
  return __builtin_amdgcn_wmma_f32_16x16x4_f32(
      false, a, false, b, (short)0, c, false, false);
}

// TDM: DMA a DIM x PCOLS f32 panel (row stride DIM floats in global) into LDS
// at byte offset lds_off, padded to PSTRIDE floats per row.
__device__ __forceinline__ void tdm_load_panel(const float* gsrc, unsigned lds_off) {
  unsigned long long ga = (unsigned long long)gsrc;
  v4u g0;
  g0.x = 1u;                                                  // count=1, user desc
  g0.y = lds_off;                                             // lds_addr (bytes)
  g0.z = (unsigned)(ga & 0xFFFFFFFFu);                        // global_addr[31:0]
  g0.w = (unsigned)((ga >> 32) & 0x01FFFFFFu) | (2u << 30);   // addr[56:32] | type=2
  v8i g1;
  // data_size=4B (2), pad_enable, pad_interval=64 dw (5), pad_amount=4 dw (3)
  g1[0] = (int)((2u << 16) | (1u << 20) | (5u << 22) | (3u << 25));
  g1[1] = (int)(((unsigned)DIM & 0xFFFFu) << 16);   // tensor_dim0[15:0]
  g1[2] = (int)(((unsigned)DIM & 0xFFFFu) << 16);   // dim0 hi=0 | tensor_dim1[15:0]
  g1[3] = (int)(((unsigned)PCOLS) << 16);           // dim1 hi=0 | tile_dim0
  g1[4] = DIM;                                      // tile_dim1=256, tile_dim2=0
  g1[5] = DIM;                                      // tensor_dim0_stride[31:0]
  g1[6] = 0;                                        // stride0 hi | dim1_stride (2D)
  g1[7] = 0;
  v4i z4 = {0, 0, 0, 0};                            // groups 2/3 unused (2D tensor)
  v8i z8 = {0, 0, 0, 0, 0, 0, 0, 0};
  __builtin_amdgcn_tensor_load_to_lds(g0, g1, z4, z4, z8, 0);
}

__global__ __launch_bounds__(256) void gated_dualproj_kernel(
    const float* __restrict__ i0, const float* __restrict__ i1,
    const float* __restrict__ W1, const float* __restrict__ W2,
    const float* __restrict__ A1, const float* __restrict__ A2,
    float* __restrict__ out)
{
  // Single LDS arena: [panel0 | panel1 | red0 | red1]
  __shared__ __align__(16) unsigned char smem[2 * PANEL_BYTES + 2 * 8192];
  float* pan0 = (float*)smem;
  float* pan1 = (float*)(smem + PANEL_BYTES);
  float* red0 = (float*)(smem + RED_OFF);          // 256 lanes x 8 row-partials
  float* red1 = red0 + 2048;

  const int lane = threadIdx.x & 31;
  const int wv   = threadIdx.x >> 5;
  const int m    = lane & 15;   // row (A) / column (B,C,D) within tile
  const int hi   = lane >> 4;   // K-pair / row-half selector
  const int row0 = (blockIdx.x * 8 + wv) * 16;

  const float* i0r = i0 + (size_t)(row0 + m) * DIM;
  const float* i1r = i1 + (size_t)(row0 + m) * DIM;

  // ---------------- Phase 1: bilinear gate scores ----------------
  float g0p[8], g1p[8];
  #pragma unroll
  for (int r = 0; r < 8; ++r) { g0p[r] = 0.f; g1p[r] = 0.f; }

  for (int pg = 0; pg < DIM / PCOLS; ++pg) {
    __syncthreads();                                // previous panel fully consumed
    if (wv == 0) {
      tdm_load_panel(A1 + (size_t)pg * PCOLS, 0);
      tdm_load_panel(A2 + (size_t)pg * PCOLS, PANEL_BYTES);
      __builtin_amdgcn_s_wait_tensorcnt(0);
    }
    __syncthreads();                                // publish panel to all waves

    for (int ntl = 0; ntl < PCOLS / 16; ++ntl) {
      const int coll = ntl * 16 + m;
      const int col  = pg * PCOLS + coll;
      v8f accA = {0.f,0.f,0.f,0.f,0.f,0.f,0.f,0.f};  // I1 @ attn1 tile
      v8f accB = {0.f,0.f,0.f,0.f,0.f,0.f,0.f,0.f};  // I0 @ attn2 tile
      #pragma unroll 4
      for (int k0 = 0; k0 < DIM; k0 += 4) {
        const int kk = k0 + 2 * hi;
        v2f a1 = *(const v2f*)(i1r + kk);
        v2f a0 = *(const v2f*)(i0r + kk);
        v2f b1; b1.x = pan0[kk * PSTRIDE + coll]; b1.y = pan0[(kk + 1) * PSTRIDE + coll];
        v2f b2; b2.x = pan1[kk * PSTRIDE + coll]; b2.y = pan1[(kk + 1) * PSTRIDE + coll];
        accA = wmma_f32(a1, b1, accA);
        accB = wmma_f32(a0, b2, accB);
      }
      // fold tile into per-row gate partials: g0[mr] += T0[mr,col] * i0[mr,col]
      #pragma unroll
      for (int r = 0; r < 8; ++r) {
        const int mr = row0 + hi * 8 + r;
        g0p[r] += accA[r] * i0[(size_t)mr * DIM + col];
        g1p[r] += accB[r] * i1[(size_t)mr * DIM + col];
      }
    }
  }

  #pragma unroll
  for (int r = 0; r < 8; ++r) {
    red0[(wv * 32 + lane) * 8 + r] = g0p[r];
    red1[(wv * 32 + lane) * 8 + r] = g1p[r];
  }
  __syncthreads();

  // lane needs the gate of its A-fragment row m = lane&15.
  // rows 0..7 accumulated by lanes 0..15 (idx=m), rows 8..15 by lanes 16..31 (idx=m-8)
  float gs0 = 0.f, gs1 = 0.f;
  {
    const int base = (m < 8) ? 0 : 16;
    const int idx  = m & 7;
    #pragma unroll
    for (int l = 0; l < 16; ++l) {
      gs0 += red0[(wv * 32 + base + l) * 8 + idx];
      gs1 += red1[(wv * 32 + base + l) * 8 + idx];
    }
  }
  const float s0 = 1.0f + 1.0f / (1.0f + __expf(-gs0));  // 1 + sigmoid
  const float s1 = 1.0f + 1.0f / (1.0f + __expf(-gs1));

  // ------- Phase 2: out = diag(s0)*I0 @ W1 + diag(s1)*I1 @ W2 -------
  for (int pg = 0; pg < DIM / PCOLS; ++pg) {
    __syncthreads();
    if (wv == 0) {
      tdm_load_panel(W1 + (size_t)pg * PCOLS, 0);
      tdm_load_panel(W2 + (size_t)pg * PCOLS, PANEL_BYTES);
      __builtin_amdgcn_s_wait_tensorcnt(0);
    }
    __syncthreads();

    v8f acc[4];
    #pragma unroll
    for (int t = 0; t < 4; ++t) acc[t] = (v8f){0.f,0.f,0.f,0.f,0.f,0.f,0.f,0.f};

    #pragma unroll 2
    for (int k0 = 0; k0 < DIM; k0 += 4) {
      const int kk = k0 + 2 * hi;
      v2f a0 = *(const v2f*)(i0r + kk); a0.x *= s0; a0.y *= s0;
      v2f a1 = *(const v2f*)(i1r + kk); a1.x *= s1; a1.y *= s1;
      #pragma unroll
      for (int t = 0; t < 4; ++t) {
        const int coll = t * 16 + m;
        v2f b1; b1.x = pan0[kk * PSTRIDE + coll]; b1.y = pan0[(kk + 1) * PSTRIDE + coll];
        v2f b2; b2.x = pan1[kk * PSTRIDE + coll]; b2.y = pan1[(kk + 1) * PSTRIDE + coll];
        acc[t] = wmma_f32(a0, b1, acc[t]);
        acc[t] = wmma_f32(a1, b2, acc[t]);
      }
    }
    #pragma unroll
    for (int t = 0; t < 4; ++t) {
      const int col = pg * PCOLS + t * 16 + m;
      #pragma unroll
      for (int r = 0; r < 8; ++r)
        out[(size_t)(row0 + hi * 8 + r) * DIM + col] = acc[t][r];
    }
  }
}

extern "C" void kernel_launch(void* const* d_in, const int* in_sizes, int n_in,
                              void* d_out, int out_size, void* d_ws, size_t ws_size,
                              hipStream_t stream) {
  const float* i0 = (const float*)d_in[0];
  const float* i1 = (const float*)d_in[1];
  const float* W1 = (const float*)d_in[2];
  const float* W2 = (const float*)d_in[3];
  const float* A1 = (const float*)d_in[4];
  const float* A2 = (const float*)d_in[5];
  float* out = (float*)d_out;

  const int nrows = in_sizes[0] / DIM;   // B*A*I = 65536
  const int blocks = nrows / 128;        // 8 waves x 16 rows per block
  gated_dualproj_kernel<<<blocks, 256, 0, stream>>>(i0, i1, W1, W2, A1, A2, out);
}